// CRFLayer_23948737642760
// MI455X (gfx1250) — compile-verified
//
#include <hip/hip_runtime.h>
#include <hip/hip_bf16.h>

// Problem constants (fixed by the reference).
#define B_ 128
#define T_ 512
#define D_ 512
#define L_ 128

typedef __attribute__((ext_vector_type(2))) float v2f;
typedef __attribute__((ext_vector_type(8))) float v8f;

#define WMMA_F32(A, Bf, C) \
    __builtin_amdgcn_wmma_f32_16x16x4_f32(false, (A), false, (Bf), (short)0, (C), false, false)

// ---------------------------------------------------------------------------
// Kernel 1: emissions[BT, L] = feats[BT, D] @ W[D, L] + b[L]
// One block = 64 rows of feats staged in LDS (stride 516 -> conflict-free
// A-fragment reads). 8 waves, each producing a 32x32 fp32 tile: 4 wmma per
// 2 A-frag + 2 B-frag loads (2x better load:wmma ratio than 16x16 tiling).
// ---------------------------------------------------------------------------
__global__ __launch_bounds__(256) void emis_gemm(const float* __restrict__ feats,
                                                 const float* __restrict__ W,
                                                 const float* __restrict__ bias,
                                                 float* __restrict__ em) {
    __shared__ float Alds[64 * 516];
    const int tid = threadIdx.x;
    const int m0  = blockIdx.x * 64;

    // Cooperative vectorized load of the 64x512 A tile (32 float4 per thread).
    const float4* f4 = (const float4*)(feats + m0 * D_);
#pragma unroll
    for (int i = 0; i < 32; ++i) {
        int idx = tid + i * 256;          // 0..8191 float4s
        int r   = idx >> 7;               // 128 float4 per row
        int c4  = idx & 127;
        float4 v = f4[r * 128 + c4];
        float* dst = &Alds[r * 516 + c4 * 4];
        dst[0] = v.x; dst[1] = v.y; dst[2] = v.z; dst[3] = v.w;
    }
    __syncthreads();

    const int wave = tid >> 5, lane = tid & 31;
    const int mpos = (wave & 1) * 32;     // M block within the 64 rows
    const int npos = (wave >> 1) * 32;    // N block within 128 cols
    const int n   = lane & 15;
    const int khi = lane >> 4;
    const int mA  = lane & 15;
    const int mb  = khi * 8;

    v8f a00 = {0.f,0.f,0.f,0.f,0.f,0.f,0.f,0.f};
    v8f a01 = a00, a10 = a00, a11 = a00;

    for (int k = 0; k < D_; k += 4) {
        v2f fa0, fa1, fb0, fb1;
        const float* Ap = &Alds[(mpos + mA) * 516 + k + 2 * khi];
        fa0.x = Ap[0];            fa0.y = Ap[1];
        fa1.x = Ap[16 * 516];     fa1.y = Ap[16 * 516 + 1];
        const float* Wp = &W[(k + 2 * khi) * L_ + npos + n];
        fb0.x = Wp[0];            fb0.y = Wp[L_];
        fb1.x = Wp[16];           fb1.y = Wp[L_ + 16];
        a00 = WMMA_F32(fa0, fb0, a00);
        a01 = WMMA_F32(fa0, fb1, a01);
        a10 = WMMA_F32(fa1, fb0, a10);
        a11 = WMMA_F32(fa1, fb1, a11);
    }

    const float bv0 = bias[npos + n];
    const float bv1 = bias[npos + 16 + n];
#pragma unroll
    for (int v = 0; v < 8; ++v) {
        int r0 = m0 + mpos + mb + v;
        int r1 = r0 + 16;
        em[r0 * L_ + npos + n]      = a00[v] + bv0;
        em[r0 * L_ + npos + 16 + n] = a01[v] + bv1;
        em[r1 * L_ + npos + n]      = a10[v] + bv0;
        em[r1 * L_ + npos + 16 + n] = a11[v] + bv1;
    }
}

// ---------------------------------------------------------------------------
// Kernel 2: CRF forward scan + gold score + final reduction.
// 8 blocks x 16 batches. Per step t (2 barriers only):
//   - em[t] loads issued early (latency hidden under shuffle-reduce + WMMA)
//   - per-wave: 2 batches, half-wave shfl_xor row-max, exp -> vb, publish marr
//   - barrier; [16,128]x[128,128] fp32-WMMA against E[p][c]=exp(trans[c][p])
//   - masked scores update: em + m + log(sum); barrier
// ---------------------------------------------------------------------------
#define SCS 132   // padded row stride (floats) for LDS matrices

__global__ __launch_bounds__(256) void crf_scan(const float* __restrict__ em,
                                                const int*   __restrict__ lengths,
                                                const int*   __restrict__ tags,
                                                const float* __restrict__ startT,
                                                const float* __restrict__ trans,
                                                const float* __restrict__ endT,
                                                float* __restrict__ out) {
    __shared__ float Bm[128 * SCS];   // Bm[p][c] = exp(trans[c][p])
    __shared__ float sc[16 * SCS];    // running scores [batch][label]
    __shared__ float vb[16 * SCS];    // exp(scores - m)
    __shared__ float marr[16];        // per-batch running max
    __shared__ float red[16][16];     // reduction scratch (init/final phases)
    __shared__ float resv[16];        // per-batch logZ, then logZ-gold

    const int tid = threadIdx.x;
    const int bg  = blockIdx.x * 16;  // first batch of this block

    // Init E matrix (transposed+exp'd transitions) and t=0 scores.
    for (int idx = tid; idx < 128 * 128; idx += 256) {
        int p = idx >> 7, c = idx & 127;
        Bm[p * SCS + c] = __expf(trans[c * 128 + p]);
    }
    for (int idx = tid; idx < 16 * 128; idx += 256) {
        int b = idx >> 7, l = idx & 127;
        sc[b * SCS + l] = em[(bg + b) * (T_ * L_) + l] + startT[l];
    }
    __syncthreads();

    const int wave = tid >> 5, lane = tid & 31;
    const int n0  = wave * 16;
    const int n   = lane & 15;
    const int khi = lane >> 4;
    const int mA  = lane & 15;
    const int mb  = khi * 8;
    const int half = lane >> 4;       // which of the wave's 2 batches
    const int j    = lane & 15;       // lane within the 16-lane half
    const int bw   = 2 * wave + half; // this half-wave's batch (0..15)
    const int rb  = tid >> 4;         // reduction batch 0..15 (init/final)
    const int rj  = tid & 15;

    int mylen[8];
#pragma unroll
    for (int v = 0; v < 8; ++v) mylen[v] = lengths[bg + mb + v];

    for (int t = 1; t < T_; ++t) {
        // Issue this step's emission loads early; latency hidden below.
        float efr[8];
#pragma unroll
        for (int v = 0; v < 8; ++v)
            efr[v] = em[((bg + mb + v) * T_ + t) * L_ + n0 + n];

        // ---- per-batch max (half-wave shfl reduce) + exp, no extra barriers ----
        float x[8];
        float pm = -3.4e38f;
#pragma unroll
        for (int i = 0; i < 8; ++i) {
            x[i] = sc[bw * SCS + j + 16 * i];
            pm = fmaxf(pm, x[i]);
        }
#pragma unroll
        for (int msk = 1; msk < 16; msk <<= 1)
            pm = fmaxf(pm, __shfl_xor(pm, msk, 32));
        if (j == 0) marr[bw] = pm;
#pragma unroll
        for (int i = 0; i < 8; ++i)
            vb[bw * SCS + j + 16 * i] = __expf(x[i] - pm);
        __syncthreads();

        // ---- [16,128] x [128,128] fp32 WMMA GEMM ----
        v8f acc = {0.f,0.f,0.f,0.f,0.f,0.f,0.f,0.f};
        for (int kk = 0; kk < 128; kk += 4) {
            v2f a, bfr;
            a.x   = vb[mA * SCS + kk + 2 * khi];
            a.y   = vb[mA * SCS + kk + 2 * khi + 1];
            bfr.x = Bm[(kk + 2 * khi) * SCS + n0 + n];
            bfr.y = Bm[(kk + 2 * khi + 1) * SCS + n0 + n];
            acc = WMMA_F32(a, bfr, acc);
        }

        // ---- masked scores update: em + m + log(sum) ----
#pragma unroll
        for (int v = 0; v < 8; ++v) {
            int bb = mb + v;
            if (t < mylen[v])
                sc[bb * SCS + n0 + n] = efr[v] + marr[bb] + __logf(acc[v]);
        }
        __syncthreads();
    }

    // ---- logZ[b] = logsumexp(scores + end_transition) ----
    float pm = -3.4e38f;
#pragma unroll
    for (int i = 0; i < 8; ++i) {
        int l = rj + 16 * i;
        pm = fmaxf(pm, sc[rb * SCS + l] + endT[l]);
    }
    red[rb][rj] = pm;
    __syncthreads();
    if (tid < 16) {
        float mm = red[tid][0];
#pragma unroll
        for (int i = 1; i < 16; ++i) mm = fmaxf(mm, red[tid][i]);
        marr[tid] = mm;
    }
    __syncthreads();
    float ps = 0.f;
#pragma unroll
    for (int i = 0; i < 8; ++i) {
        int l = rj + 16 * i;
        ps += __expf(sc[rb * SCS + l] + endT[l] - marr[rb]);
    }
    red[rb][rj] = ps;
    __syncthreads();
    if (tid < 16) {
        float s = 0.f;
#pragma unroll
        for (int i = 0; i < 16; ++i) s += red[tid][i];
        resv[tid] = marr[tid] + __logf(s);   // logZ
    }
    __syncthreads();

    // ---- gold path score (parallel over t, 16 threads per batch) ----
    {
        const int len  = lengths[bg + rb];
        const int base = (bg + rb) * T_;
        float pg = 0.f;
        for (int t = 1 + rj; t < T_; t += 16) {
            if (t < len) {
                int tg = tags[base + t];
                int pv = tags[base + t - 1];
                pg += trans[tg * L_ + pv] + em[(base + t) * L_ + tg];
            }
        }
        if (rj == 0) {
            int t0 = tags[base];
            pg += startT[t0] + em[base * L_ + t0];
            pg += endT[tags[base + len - 1]];
        }
        red[rb][rj] = pg;
    }
    __syncthreads();
    if (tid < 16) {
        float g = 0.f;
#pragma unroll
        for (int i = 0; i < 16; ++i) g += red[tid][i];
        resv[tid] -= g;                       // logZ - gold
    }
    __syncthreads();
    if (tid == 0) {
        float s = 0.f;
#pragma unroll
        for (int i = 0; i < 16; ++i) s += resv[i];
        atomicAdd(out, s);
    }
}

// ---------------------------------------------------------------------------
extern "C" void kernel_launch(void* const* d_in, const int* in_sizes, int n_in,
                              void* d_out, int out_size, void* d_ws, size_t ws_size,
                              hipStream_t stream) {
    (void)in_sizes; (void)n_in; (void)out_size; (void)ws_size;
    const float* feats   = (const float*)d_in[0];
    const int*   lengths = (const int*)d_in[1];
    const int*   tags    = (const int*)d_in[2];
    const float* W       = (const float*)d_in[3];
    const float* bias    = (const float*)d_in[4];
    const float* startT  = (const float*)d_in[5];
    const float* trans   = (const float*)d_in[6];
    const float* endT    = (const float*)d_in[7];

    float* em = (float*)d_ws;   // 65536 x 128 fp32 emissions (32 MB)

    emis_gemm<<<(B_ * T_) / 64, 256, 0, stream>>>(feats, W, bias, em);
    hipMemsetAsync(d_out, 0, sizeof(float), stream);
    crf_scan<<<B_ / 16, 256, 0, stream>>>(em, lengths, tags, startT, trans, endT,
                                          (float*)d_out);
}